// MoE_Layer_79912161509423
// MI455X (gfx1250) — compile-verified
//
#include <hip/hip_runtime.h>
#include <math.h>

// ---------------- problem constants (match reference) ----------------
#define T_TOK 8192      // B*S = 4*2048
#define D_DIM 1024
#define E_NUM 8
#define H_DIM 4096
#define MT    64        // gathered tokens per block tile (weight reuse lever)
#define HC    128       // H chunk per fused iteration
#define XPAD  4         // xs row padding (stride 1028 -> conflict-free A reads)
#define HPAD  4         // hs row padding (stride 132)
#define RT    256       // router block threads
#define ET    512       // expert block threads = 16 wave32

typedef float v2f __attribute__((ext_vector_type(2)));
typedef float v8f __attribute__((ext_vector_type(8)));

// ---------------------------------------------------------------------
// Kernel 0: zero output + per-expert counters
// ---------------------------------------------------------------------
__global__ void moe_zero_kernel(float* __restrict__ out, int n, int* __restrict__ counts) {
    int i = blockIdx.x * blockDim.x + threadIdx.x;
    if (i < E_NUM) counts[i] = 0;
    for (int j = i; j < n; j += gridDim.x * blockDim.x) out[j] = 0.0f;
}

// ---------------------------------------------------------------------
// Kernel 1: router — one wave per token, top-2 of logits, gather lists
// ---------------------------------------------------------------------
__global__ void moe_router_kernel(const float* __restrict__ x,
                                  const float* __restrict__ rw,
                                  const float* __restrict__ rb,
                                  int* __restrict__ counts,
                                  int* __restrict__ idxl) {
    const int wave = threadIdx.x >> 5;
    const int lane = threadIdx.x & 31;
    const int t = blockIdx.x * (RT / 32) + wave;
    if (t >= T_TOK) return;

    float acc[E_NUM];
#pragma unroll
    for (int e = 0; e < E_NUM; ++e) acc[e] = 0.0f;

    const float* xr = x + (size_t)t * D_DIM;
    for (int d = lane; d < D_DIM; d += 32) {
        float xv = xr[d];
        const float* w = rw + (size_t)d * E_NUM;
#pragma unroll
        for (int e = 0; e < E_NUM; ++e) acc[e] += xv * w[e];
    }
#pragma unroll
    for (int e = 0; e < E_NUM; ++e) {
#pragma unroll
        for (int off = 16; off > 0; off >>= 1)
            acc[e] += __shfl_xor(acc[e], off, 32);
    }
    if (lane == 0) {
        int e1 = 0, e2 = -1;
        float l1 = acc[0] + rb[0], l2 = -INFINITY;
#pragma unroll
        for (int e = 1; e < E_NUM; ++e) {
            float l = acc[e] + rb[e];
            if (l > l1) { l2 = l1; e2 = e1; l1 = l; e1 = e; }
            else if (l > l2) { l2 = l; e2 = e; }
        }
        int p1 = atomicAdd(&counts[e1], 1);
        idxl[e1 * T_TOK + p1] = t;
        int p2 = atomicAdd(&counts[e2], 1);
        idxl[e2 * T_TOK + p2] = t;
    }
}

// ---------------------------------------------------------------------
// Kernel 2: fused expert FFN on gathered tokens.
//   block = 16 waves, tile = 64 tokens. x rows staged ONCE into LDS via
//   global_load_async_to_lds_b128 (ASYNCcnt). For each H-chunk of 128:
//   GEMM1 (wmma f32 16x16x4, K=1024, A from LDS) + b1 + exact GELU -> LDS
//   GEMM2 accumulates y[64,1024] in registers from LDS h + global w2.
//   Epilogue: + b2, atomicAdd into scattered token rows of out.
// ---------------------------------------------------------------------
__global__ __launch_bounds__(ET)
void moe_expert_kernel(const float* __restrict__ x,
                       const float* __restrict__ w1,
                       const float* __restrict__ b1,
                       const float* __restrict__ w2,
                       const float* __restrict__ b2,
                       const int* __restrict__ counts,
                       const int* __restrict__ idxl,
                       float* __restrict__ out) {
    const int e    = blockIdx.y;
    const int tile = blockIdx.x;
    const int cnt  = counts[e];
    const int row0 = tile * MT;
    if (row0 >= cnt) return;

    __shared__ float xs[MT][D_DIM + XPAD];   // 64 x 1028 f32 ~= 263 KB
    __shared__ float hs[MT][HC + HPAD];      // 64 x 132  f32 ~=  34 KB
    __shared__ int   toks[MT];

    const int tid  = threadIdx.x;
    const int wave = tid >> 5;
    const int lane = tid & 31;
    const int lo   = lane & 15;
    const int hi   = lane >> 4;
    const int mgrp = wave >> 3;      // 0..1 -> token rows [32*mgrp, 32*mgrp+32)
    const int wn   = wave & 7;       // GEMM1 N-tile / GEMM2 N-slice

    if (tid < MT) {
        int r = row0 + tid;
        toks[tid] = (r < cnt) ? idxl[e * T_TOK + r] : -1;
    }
    __syncthreads();

    // ---- stage gathered x rows into LDS (async DMA to LDS, b128) ----
    // 64 rows x 256 float4 = 16384 float4 slots; 512 threads -> 32 each.
    for (int it = 0; it < (MT * (D_DIM / 4)) / ET; ++it) {
        int s   = tid + it * ET;
        int r   = s >> 8;            // /(D_DIM/4)
        int c4  = s & 255;
        int tk  = toks[r];
        unsigned lds = (unsigned)(size_t)&xs[r][c4 * 4];
        if (tk >= 0) {
            const float* gp = x + (size_t)tk * D_DIM + c4 * 4;
            asm volatile("global_load_async_to_lds_b128 %0, %1, off"
                         :: "v"(lds), "v"(gp) : "memory");
        } else {
            xs[r][c4 * 4 + 0] = 0.0f;
            xs[r][c4 * 4 + 1] = 0.0f;
            xs[r][c4 * 4 + 2] = 0.0f;
            xs[r][c4 * 4 + 3] = 0.0f;
        }
    }
    asm volatile("s_wait_asynccnt 0" ::: "memory");
    __syncthreads();

    const float* w1e = w1 + (size_t)e * D_DIM * H_DIM;
    const float* w2e = w2 + (size_t)e * H_DIM * D_DIM;
    const float* b1e = b1 + (size_t)e * H_DIM;
    const float* b2e = b2 + (size_t)e * D_DIM;

    const int ar0 = 32 * mgrp + lo;       // A-fragment rows for this wave
    const int ar1 = 32 * mgrp + 16 + lo;

    v8f yacc[2][8];
#pragma unroll
    for (int m = 0; m < 2; ++m)
#pragma unroll
        for (int n = 0; n < 8; ++n) { v8f z = {}; yacc[m][n] = z; }

    for (int hcb = 0; hcb < H_DIM; hcb += HC) {
        // ---- GEMM1: wave computes h tile rows [32m,32m+32), cols wn*16 ----
        const int gc = hcb + wn * 16 + lo;      // global h column (B/bias)
        v8f h0 = {}, h1 = {};
        for (int k = 0; k < D_DIM; k += 4) {
            const int k2 = k + 2 * hi;
            v2f a0 = *(const v2f*)&xs[ar0][k2];
            v2f a1 = *(const v2f*)&xs[ar1][k2];
            v2f b;
            b.x = w1e[(size_t)k2 * H_DIM + gc];
            b.y = w1e[(size_t)(k2 + 1) * H_DIM + gc];
            h0 = __builtin_amdgcn_wmma_f32_16x16x4_f32(false, a0, false, b,
                                                       (short)0, h0, false, false);
            h1 = __builtin_amdgcn_wmma_f32_16x16x4_f32(false, a1, false, b,
                                                       (short)0, h1, false, false);
        }
        // bias + exact GELU, stage chunk to LDS (C layout: vgpr i -> row hi*8+i)
        const float bias = b1e[gc];
        const int ccol = wn * 16 + lo;
#pragma unroll
        for (int i = 0; i < 8; ++i) {
            float v0 = h0[i] + bias;
            v0 = 0.5f * v0 * (1.0f + erff(v0 * 0.70710678118f));
            hs[32 * mgrp + hi * 8 + i][ccol] = v0;
            float v1 = h1[i] + bias;
            v1 = 0.5f * v1 * (1.0f + erff(v1 * 0.70710678118f));
            hs[32 * mgrp + 16 + hi * 8 + i][ccol] = v1;
        }
        __syncthreads();

        // ---- GEMM2: wave accumulates y rows [32m,+32), cols [wn*128,+128) ----
        for (int kk = 0; kk < HC; kk += 4) {
            const int k2 = kk + 2 * hi;
            v2f a0 = *(const v2f*)&hs[ar0][k2];
            v2f a1 = *(const v2f*)&hs[ar1][k2];
            const size_t rb0 = (size_t)(hcb + k2) * D_DIM;
#pragma unroll
            for (int nt = 0; nt < 8; ++nt) {
                const int c = wn * 128 + nt * 16 + lo;
                v2f b;
                b.x = w2e[rb0 + c];
                b.y = w2e[rb0 + D_DIM + c];
                yacc[0][nt] = __builtin_amdgcn_wmma_f32_16x16x4_f32(
                    false, a0, false, b, (short)0, yacc[0][nt], false, false);
                yacc[1][nt] = __builtin_amdgcn_wmma_f32_16x16x4_f32(
                    false, a1, false, b, (short)0, yacc[1][nt], false, false);
            }
        }
        __syncthreads();   // hs will be overwritten next chunk
    }

    // ---- epilogue: +b2, scatter-accumulate into out (fp32 atomics) ----
#pragma unroll
    for (int nt = 0; nt < 8; ++nt) {
        const int c = wn * 128 + nt * 16 + lo;
        const float bias2 = b2e[c];
#pragma unroll
        for (int i = 0; i < 8; ++i) {
            const int r0 = 32 * mgrp + hi * 8 + i;
            const int tk0 = toks[r0];
            if (tk0 >= 0)
                atomicAdd(out + (size_t)tk0 * D_DIM + c, yacc[0][nt][i] + bias2);
            const int r1 = 32 * mgrp + 16 + hi * 8 + i;
            const int tk1 = toks[r1];
            if (tk1 >= 0)
                atomicAdd(out + (size_t)tk1 * D_DIM + c, yacc[1][nt][i] + bias2);
        }
    }
}

// ---------------------------------------------------------------------
extern "C" void kernel_launch(void* const* d_in, const int* in_sizes, int n_in,
                              void* d_out, int out_size, void* d_ws, size_t ws_size,
                              hipStream_t stream) {
    const float* x   = (const float*)d_in[0];
    const float* rw  = (const float*)d_in[1];
    const float* rb  = (const float*)d_in[2];
    const float* w1  = (const float*)d_in[3];
    const float* b1  = (const float*)d_in[4];
    const float* w2  = (const float*)d_in[5];
    const float* b2  = (const float*)d_in[6];
    float* out = (float*)d_out;

    // workspace layout: [0,32) expert counters, [256, 256 + E*T*4) gather lists
    int* counts = (int*)d_ws;
    int* idxl   = (int*)((char*)d_ws + 256);

    const int n_out = T_TOK * D_DIM;
    moe_zero_kernel<<<4096, RT, 0, stream>>>(out, n_out, counts);

    moe_router_kernel<<<T_TOK / (RT / 32), RT, 0, stream>>>(
        x, rw, rb, counts, idxl);

    dim3 grid(T_TOK / MT, E_NUM);   // worst case: all tokens on one expert
    moe_expert_kernel<<<grid, ET, 0, stream>>>(
        x, w1, b1, w2, b2, counts, idxl, out);
}